// DiagGraphSAGENet_25460566130863
// MI455X (gfx1250) — compile-verified
//
#include <hip/hip_runtime.h>
#include <hip/hip_bf16.h>
#include <math.h>

// ---------------------------------------------------------------------------
// DiagGraphSAGENet on MI455X (gfx1250)
//   agg = segment_sum(x[src], dst)                    (HBM/atomic bound)
//   loc   = clip(agg@Wl1^T + bl1 + x@Wr1^T, -100,100) (f32 WMMA)
//   scale = min(softplus(agg@Wl2^T + bl2 + x@Wr2^T)+1e-3, 100)
// ---------------------------------------------------------------------------

#define N_NODES 50000
#define N_EDGES 800000
#define IN_DIM  512
#define OUT_DIM 512

typedef __attribute__((ext_vector_type(2))) float v2f;
typedef __attribute__((ext_vector_type(8))) float v8f;

// ---------------------------------------------------------------------------
// Kernel 1: zero the aggregation workspace (float4 grid-stride)
// ---------------------------------------------------------------------------
__global__ void zero_agg_kernel(float4* __restrict__ agg4, int n4) {
    int i = blockIdx.x * blockDim.x + threadIdx.x;
    int stride = gridDim.x * blockDim.x;
    float4 z = make_float4(0.f, 0.f, 0.f, 0.f);
    for (; i < n4; i += stride) agg4[i] = z;
}

// ---------------------------------------------------------------------------
// Kernel 2: scatter-add of source rows into destination rows.
// One wave32 per edge: lane c handles feature dims c, c+32, ... (coalesced).
// Lowers to global_atomic_add_f32 (L2 atomic units, no return value).
// ---------------------------------------------------------------------------
__global__ void edge_scatter_kernel(const float* __restrict__ x,
                                    const long long* __restrict__ edge_index,
                                    float* __restrict__ agg,
                                    int n_edges) {
    const int lane         = threadIdx.x & 31;
    const int wave_in_bl   = threadIdx.x >> 5;
    const int waves_per_bl = blockDim.x >> 5;
    long long e = (long long)blockIdx.x * waves_per_bl + wave_in_bl;
    const long long e_stride = (long long)gridDim.x * waves_per_bl;

    for (; e < n_edges; e += e_stride) {
        const long long s = edge_index[e];            // src row
        const long long d = edge_index[n_edges + e];  // dst row
        const float* __restrict__ xs = x   + s * IN_DIM;
        float*       __restrict__ ad = agg + d * IN_DIM;
        #pragma unroll
        for (int c = lane; c < IN_DIM; c += 32) {
            atomicAdd(&ad[c], xs[c]);
        }
    }
}

// ---------------------------------------------------------------------------
// Kernel 3: fused dual GEMM + epilogues via V_WMMA_F32_16X16X4_F32.
//
// Block = 128 threads = 4 waves sharing one 32-row M tile; each wave owns a
// 16-col N tile (adjacent waves -> adjacent weight rows, shared A rows in L0).
// Each wave accumulates FOUR 16x16 tiles: {loc, scale} x {M sub-tile 0, 1}.
// Per K-step (K=4): 8 global_load_b64 feed 8 v_wmma_f32_16x16x4_f32 (1:1),
// vs 6:4 in the single-sub-tile version — weight fragments are reused by both
// M sub-tiles.
//
// Fragment addressing (ISA 7.12.2, 32-bit operands):
//   A 16x4 : lane&15 = M row; VGPR pair supplies K = kk+2*(lane>>4) .. +1
//   B 4x16 : lane&15 = N col; B[k][n] = W[n][k] -> contiguous float2 of W row
//   C/D    : VGPR r holds (M = r + 8*(lane>>4), N = lane&15)
//
// Row tail (50000 = 1562*32 + 16): second sub-tile's load pointers are clamped
// to the first sub-tile (in-bounds, results discarded); stores are guarded by
// a wave-uniform flag so the K-loop stays branch-free with EXEC all-1s.
// ---------------------------------------------------------------------------
__global__ void
__launch_bounds__(128)
sage_wmma_kernel(const float* __restrict__ x,
                 const float* __restrict__ agg,
                 const float* __restrict__ Wl1, const float* __restrict__ bl1,
                 const float* __restrict__ Wr1,
                 const float* __restrict__ Wl2, const float* __restrict__ bl2,
                 const float* __restrict__ Wr2,
                 float* __restrict__ out_loc,
                 float* __restrict__ out_scale) {
    const int lane = threadIdx.x & 31;   // wave32 lane
    const int wave = threadIdx.x >> 5;   // 0..3: N sub-tile within block
    const int half = lane >> 4;          // 0: lanes 0-15, 1: lanes 16-31
    const int lm   = lane & 15;          // A: M row / B: N col

    const int col0 = (blockIdx.x * 4 + wave) * 16;  // N tile base (0..496)
    const int row0 = blockIdx.y * 32;               // M tile base (sub-tile 0)
    const int row1 = row0 + 16;                     // M sub-tile 1
    const bool hasB = (row1 < N_NODES);             // wave-uniform tail flag

    // A-fragment row pointers (sub-tile 1 clamped for safe tail loads).
    const float* aggRow0 = agg + (size_t)(row0 + lm) * IN_DIM;
    const float* xRow0   = x   + (size_t)(row0 + lm) * IN_DIM;
    const size_t r1s     = (size_t)((hasB ? row1 : row0) + lm);
    const float* aggRow1 = agg + r1s * IN_DIM;
    const float* xRow1   = x   + r1s * IN_DIM;

    // B-fragment pointers: weight row (col0+lm) since GEMM is  @ W^T.
    const float* w1lRow = Wl1 + (size_t)(col0 + lm) * IN_DIM;
    const float* w1rRow = Wr1 + (size_t)(col0 + lm) * IN_DIM;
    const float* w2lRow = Wl2 + (size_t)(col0 + lm) * IN_DIM;
    const float* w2rRow = Wr2 + (size_t)(col0 + lm) * IN_DIM;

    v8f accL0 = {};   // loc,   rows row0..row0+15
    v8f accS0 = {};   // scale, rows row0..row0+15
    v8f accL1 = {};   // loc,   rows row1..row1+15
    v8f accS1 = {};   // scale, rows row1..row1+15

    #pragma unroll 2
    for (int kk = 0; kk < IN_DIM; kk += 4) {
        const int kh = kk + 2 * half;   // this lane-half supplies K = kh, kh+1

        // 4 weight fragments, shared by both M sub-tiles.
        v2f b_l1 = *(const v2f*)(w1lRow + kh);
        v2f b_r1 = *(const v2f*)(w1rRow + kh);
        v2f b_l2 = *(const v2f*)(w2lRow + kh);
        v2f b_r2 = *(const v2f*)(w2rRow + kh);
        // 4 activation fragments (agg & x for both sub-tiles).
        v2f a0a  = *(const v2f*)(aggRow0 + kh);
        v2f a0x  = *(const v2f*)(xRow0   + kh);
        v2f a1a  = *(const v2f*)(aggRow1 + kh);
        v2f a1x  = *(const v2f*)(xRow1   + kh);

        accL0 = __builtin_amdgcn_wmma_f32_16x16x4_f32(
            false, a0a, false, b_l1, (short)0, accL0, false, false);
        accL0 = __builtin_amdgcn_wmma_f32_16x16x4_f32(
            false, a0x, false, b_r1, (short)0, accL0, false, false);
        accS0 = __builtin_amdgcn_wmma_f32_16x16x4_f32(
            false, a0a, false, b_l2, (short)0, accS0, false, false);
        accS0 = __builtin_amdgcn_wmma_f32_16x16x4_f32(
            false, a0x, false, b_r2, (short)0, accS0, false, false);
        accL1 = __builtin_amdgcn_wmma_f32_16x16x4_f32(
            false, a1a, false, b_l1, (short)0, accL1, false, false);
        accL1 = __builtin_amdgcn_wmma_f32_16x16x4_f32(
            false, a1x, false, b_r1, (short)0, accL1, false, false);
        accS1 = __builtin_amdgcn_wmma_f32_16x16x4_f32(
            false, a1a, false, b_l2, (short)0, accS1, false, false);
        accS1 = __builtin_amdgcn_wmma_f32_16x16x4_f32(
            false, a1x, false, b_r2, (short)0, accS1, false, false);
    }

    // ---------------- epilogue ----------------
    const int   col = col0 + lm;
    const float b1  = bl1[col];
    const float b2  = bl2[col];

    #pragma unroll
    for (int r = 0; r < 8; ++r) {
        const int rowA = row0 + r + 8 * half;
        const size_t idxA = (size_t)rowA * OUT_DIM + col;

        float vloc = accL0[r] + b1;                       // clip(v,-100,100)
        out_loc[idxA] = fminf(fmaxf(vloc, -100.0f), 100.0f);

        float vsc = accS0[r] + b2;                        // softplus, stable
        float sp  = (vsc > 20.0f) ? vsc : log1pf(__expf(vsc));
        out_scale[idxA] = fminf(sp + 0.001f, 100.0f);

        if (hasB) {
            const int rowB = row1 + r + 8 * half;
            const size_t idxB = (size_t)rowB * OUT_DIM + col;

            float vloc1 = accL1[r] + b1;
            out_loc[idxB] = fminf(fmaxf(vloc1, -100.0f), 100.0f);

            float vsc1 = accS1[r] + b2;
            float sp1  = (vsc1 > 20.0f) ? vsc1 : log1pf(__expf(vsc1));
            out_scale[idxB] = fminf(sp1 + 0.001f, 100.0f);
        }
    }
}

// ---------------------------------------------------------------------------
// Launcher
// ---------------------------------------------------------------------------
extern "C" void kernel_launch(void* const* d_in, const int* in_sizes, int n_in,
                              void* d_out, int out_size, void* d_ws, size_t ws_size,
                              hipStream_t stream) {
    const float*     x   = (const float*)d_in[0];
    const long long* ei  = (const long long*)d_in[1];  // [2, N_EDGES] int64
    const float*     Wl1 = (const float*)d_in[2];
    const float*     bl1 = (const float*)d_in[3];
    const float*     Wr1 = (const float*)d_in[4];
    const float*     Wl2 = (const float*)d_in[5];
    const float*     bl2 = (const float*)d_in[6];
    const float*     Wr2 = (const float*)d_in[7];

    const int n_edges = in_sizes[1] / 2;

    float* agg = (float*)d_ws;                         // N_NODES*IN_DIM floats
    float* out_loc   = (float*)d_out;                  // [N_NODES, OUT_DIM]
    float* out_scale = out_loc + (size_t)N_NODES * OUT_DIM;

    // 1) zero agg
    {
        const int n4 = (N_NODES * IN_DIM) / 4;
        zero_agg_kernel<<<4096, 256, 0, stream>>>((float4*)agg, n4);
    }

    // 2) scatter-add edges (one wave per edge, grid-stride)
    edge_scatter_kernel<<<4096, 256, 0, stream>>>(x, ei, agg, n_edges);

    // 3) fused WMMA GEMMs + epilogues
    {
        dim3 grid(OUT_DIM / 64,                 // 8:  4 N-tiles per block
                  (N_NODES + 31) / 32);         // 1563: 32-row M tiles
        sage_wmma_kernel<<<grid, 128, 0, stream>>>(
            x, agg, Wl1, bl1, Wr1, Wl2, bl2, Wr2, out_loc, out_scale);
    }
}